// DepthAugmentedBEVLifter_12051678233323
// MI455X (gfx1250) — compile-verified
//
#include <hip/hip_runtime.h>
#include <hip/hip_bf16.h>

// CDNA5 / gfx1250 implementation of DepthAugmentedBEVLifter.
// Dense conv work on V_WMMA_F32_16X16X4_F32 (fp32-exact matrix path, wave32).
// Round-2 change: M-direction register blocking (MT m-tiles per wave) so the
// streamed B operand / im2col gather is loaded once and fed to MT back-to-back
// WMMAs -> 4x less activation traffic, 4x fewer address-VALU ops per FLOP.

typedef __attribute__((ext_vector_type(2))) float v2f;
typedef __attribute__((ext_vector_type(8))) float v8f;

#define WPB 4  // waves per block (blockDim = 128)

__device__ __forceinline__ v8f wmma_f32_k4(v2f a, v2f b, v8f c) {
  // D = A(16x4,f32) * B(4x16,f32) + C(16x16,f32)
  return __builtin_amdgcn_wmma_f32_16x16x4_f32(
      /*neg_a=*/false, a, /*neg_b=*/false, b,
      /*c_mod=*/(short)0, c, /*reuse_a=*/false, /*reuse_b=*/false);
}

// ---------------------------------------------------------------------------
// Generic 1x1-conv GEMM:  Y[b] (MxN) = post( A(MxK) * X[b](KxN) )
// MT: m-tiles (of 16 rows) per wave.  AFF: 0 none, 1 BN scale/shift, 2 bias.
// ---------------------------------------------------------------------------
template <int MT, int AFF, bool RELU, bool ACCUM>
__global__ void gemm16_kernel(const float* __restrict__ A, int lda,
                              const float* __restrict__ X, long long xbstride, int ldx,
                              float* __restrict__ Y, long long ybstride, int ldy,
                              const float* __restrict__ scale,
                              const float* __restrict__ shift,
                              int M, int K, int N) {
  const int wave = threadIdx.x >> 5;
  const int lane = threadIdx.x & 31;
  const int b = blockIdx.z;
  const int nt = (blockIdx.x * WPB + wave) * 16;
  const int mt0 = blockIdx.y * 16 * MT;
  if (nt >= N) return;  // wave-uniform: EXEC stays all-ones for WMMA
  const float* Xb = X + (long long)b * xbstride;
  float* Yb = Y + (long long)b * ybstride;
  const int half = lane >> 4;   // which K-pair this lane holds
  const int l16 = lane & 15;
  const int ncol = nt + l16;
  const int nc = (ncol < N) ? ncol : (N - 1);
  const float* Arow[MT];
#pragma unroll
  for (int i = 0; i < MT; ++i) {
    int mrow = mt0 + i * 16 + l16;
    int mr = (mrow < M) ? mrow : (M - 1);  // clamp; stores are guarded
    Arow[i] = A + (long long)mr * lda + 2 * half;
  }
  const float* Xcol = Xb + nc;
  v8f acc[MT];
#pragma unroll
  for (int i = 0; i < MT; ++i) acc[i] = (v8f){0.f, 0.f, 0.f, 0.f, 0.f, 0.f, 0.f, 0.f};
  for (int k0 = 0; k0 < K; k0 += 4) {  // K is always a multiple of 4 here
    v2f bf;
    const float* xp = Xcol + (long long)(k0 + 2 * half) * ldx;
    bf.x = xp[0];
    bf.y = xp[ldx];
    __builtin_prefetch(Xcol + (long long)(k0 + 4 + 2 * half) * ldx, 0, 0);
#pragma unroll
    for (int i = 0; i < MT; ++i) {
      v2f a;
      a.x = Arow[i][k0];
      a.y = Arow[i][k0 + 1];
      acc[i] = wmma_f32_k4(a, bf, acc[i]);
    }
  }
  if (ncol < N) {
#pragma unroll
    for (int i = 0; i < MT; ++i) {
#pragma unroll
      for (int r = 0; r < 8; ++r) {
        int row = mt0 + i * 16 + r + 8 * half;  // documented 16x16 f32 C/D layout
        if (row < M) {
          float v = acc[i][r];
          if (AFF == 1) v = v * scale[row] + shift[row];
          else if (AFF == 2) v = v + shift[row];
          if (RELU) v = fmaxf(v, 0.0f);
          long long oi = (long long)row * ldy + ncol;
          if (ACCUM) Yb[oi] += v; else Yb[oi] = v;
        }
      }
    }
  }
}

// ---------------------------------------------------------------------------
// 3x3 SAME conv (optionally grouped) as im2col-on-the-fly WMMA GEMM.
// W: (groups*Mpg, Cpg, 3, 3) OIHW.  X[b]: (groups*Cpg, H, Wd).
// The im2col gather is done ONCE per K-step and shared by MT m-tiles.
// ---------------------------------------------------------------------------
template <int MT, bool RELU>
__global__ void conv3x3_kernel(const float* __restrict__ W,
                               const float* __restrict__ X, long long xbstride,
                               float* __restrict__ Y, long long ybstride,
                               const float* __restrict__ scale,
                               const float* __restrict__ shift,
                               int H, int Wd, int Cpg, int Mpg, int groups,
                               int yChanOff) {
  const int N = H * Wd;
  const int wave = threadIdx.x >> 5;
  const int lane = threadIdx.x & 31;
  const int b = blockIdx.z;
  const int nt = (blockIdx.x * WPB + wave) * 16;
  if (nt >= N) return;  // wave-uniform
  const int mtpg = (Mpg + 16 * MT - 1) / (16 * MT);
  const int g = blockIdx.y / mtpg;
  const int mt0 = (blockIdx.y % mtpg) * 16 * MT;
  const int half = lane >> 4;
  const int l16 = lane & 15;
  const int ncol = nt + l16;
  const int nc = (ncol < N) ? ncol : (N - 1);
  const int py = nc / Wd, px = nc % Wd;
  const float* Wrow[MT];
#pragma unroll
  for (int i = 0; i < MT; ++i) {
    int mrow = mt0 + i * 16 + l16;
    int mr = (mrow < Mpg) ? mrow : (Mpg - 1);
    Wrow[i] = W + (long long)(g * Mpg + mr) * Cpg * 9 + 2 * half;
  }
  const float* Xb = X + (long long)b * xbstride + (long long)(g * Cpg) * N;
  const int K = Cpg * 9;  // 72 or 1728, both multiples of 4
  v8f acc[MT];
#pragma unroll
  for (int i = 0; i < MT; ++i) acc[i] = (v8f){0.f, 0.f, 0.f, 0.f, 0.f, 0.f, 0.f, 0.f};
  for (int k0 = 0; k0 < K; k0 += 4) {
    const int ka = k0 + 2 * half;
    v2f bf;
#pragma unroll
    for (int t = 0; t < 2; ++t) {
      int k = ka + t;
      int ci = k / 9;
      int r = k - ci * 9;
      int dy = r / 3 - 1;
      int dx = r - (r / 3) * 3 - 1;
      int yy = py + dy, xx = px + dx;
      float xv = 0.0f;
      if (yy >= 0 && yy < H && xx >= 0 && xx < Wd)
        xv = Xb[(long long)ci * N + yy * Wd + xx];
      if (t == 0) bf.x = xv; else bf.y = xv;
    }
#pragma unroll
    for (int i = 0; i < MT; ++i) {
      v2f a;
      a.x = Wrow[i][k0];
      a.y = Wrow[i][k0 + 1];
      acc[i] = wmma_f32_k4(a, bf, acc[i]);
    }
  }
  if (ncol < N) {
#pragma unroll
    for (int i = 0; i < MT; ++i) {
#pragma unroll
      for (int r = 0; r < 8; ++r) {
        int row = mt0 + i * 16 + r + 8 * half;
        if (row < Mpg) {
          int och = g * Mpg + row;
          float v = acc[i][r] * scale[och] + shift[och];
          if (RELU) v = fmaxf(v, 0.0f);
          Y[(long long)b * ybstride + (long long)(yChanOff + och) * N + ncol] = v;
        }
      }
    }
  }
}

// Softmax(10*logits) over 64 depth channels -> expected depth per pixel.
__global__ void depth_softmax_kernel(const float* __restrict__ cat, long long cstride,
                                     float* __restrict__ dmap, int N) {
  int b = blockIdx.y;
  int n = blockIdx.x * blockDim.x + threadIdx.x;
  if (n >= N) return;
  const float* L = cat + (long long)b * cstride + n;
  const float lstep = logf(35.0f) / 63.0f;  // bins = exp(linspace(0, ln 35, 64))
  float mx = -1e30f;
  for (int c = 0; c < 64; ++c) mx = fmaxf(mx, 10.0f * L[(long long)c * N]);
  float se = 0.f, sd = 0.f;
  for (int c = 0; c < 64; ++c) {
    float e = expf(10.0f * L[(long long)c * N] - mx);
    se += e;
    sd += e * expf(lstep * (float)c);
  }
  dmap[(long long)b * N + n] = sd / se;
}

// confidence head: sigmoid( w2 . hidden + b2 )
__global__ void conf_kernel(const float* __restrict__ hid, long long hstride,
                            const float* __restrict__ w, const float* __restrict__ bias,
                            float* __restrict__ conf, int N) {
  int b = blockIdx.y;
  int n = blockIdx.x * blockDim.x + threadIdx.x;
  if (n >= N) return;
  const float* Hp = hid + (long long)b * hstride + n;
  float s = bias[0];
  for (int c = 0; c < 64; ++c) s += w[c] * Hp[(long long)c * N];
  conf[(long long)b * N + n] = 1.0f / (1.0f + expf(-s));
}

// Unproject + BEV scatter-add (fp32 global atomics through L2).
__global__ void scatter_kernel(const float* __restrict__ cat, long long cstride,
                               const float* __restrict__ conf,
                               const float* __restrict__ dmap,
                               const float* __restrict__ kinv,
                               const float* __restrict__ T,
                               float* __restrict__ bev, int H, int Wd) {
  const int N = H * Wd;
  int b = blockIdx.y;
  int n = blockIdx.x * blockDim.x + threadIdx.x;
  if (n >= N) return;
  float d = dmap[(long long)b * N + n];
  float px = (float)(n % Wd), py = (float)(n / Wd);
  const float* Ki = kinv + b * 9;
  float cx = d * (Ki[0] * px + Ki[1] * py + Ki[2]);
  float cy = d * (Ki[3] * px + Ki[4] * py + Ki[5]);
  float cz = d * (Ki[6] * px + Ki[7] * py + Ki[8]);
  const float* Tb = T + b * 16;
  float ex = Tb[0] * cx + Tb[1] * cy + Tb[2] * cz + Tb[3];
  float ey = Tb[4] * cx + Tb[5] * cy + Tb[6] * cz + Tb[7];
  int bx = (int)truncf(ex / 0.8f + 64.0f);
  int by = (int)truncf(ey / 0.8f + 64.0f);
  if ((unsigned)bx >= 128u || (unsigned)by >= 128u) return;
  float wgt = conf[(long long)b * N + n];
  const float* red = cat + (long long)b * cstride + 64LL * N + n;  // reduced feats
  float* out = bev + (long long)b * 128 * 16384 + by * 128 + bx;
  for (int c = 0; c < 128; ++c)
    atomicAdd(out + (long long)c * 16384, red[(long long)c * N] * wgt);
}

// Fold eval-mode BN into scale/shift.
__global__ void fold_bn_kernel(const float* __restrict__ g, const float* __restrict__ bb,
                               const float* __restrict__ m, const float* __restrict__ v,
                               float* __restrict__ scale, float* __restrict__ shift, int C) {
  int i = blockIdx.x * blockDim.x + threadIdx.x;
  if (i < C) {
    float s = g[i] / sqrtf(v[i] + 1e-5f);
    scale[i] = s;
    shift[i] = bb[i] - m[i] * s;
  }
}

// fused BN+ReLU, in place, channel-major (C, Np) per batch.
__global__ void bnrelu_kernel(float* __restrict__ X, const float* __restrict__ scale,
                              const float* __restrict__ shift, int C, int Np) {
  int b = blockIdx.z;
  int ch = blockIdx.y;
  long long base = ((long long)b * C + ch) * Np;
  float s = scale[ch], sh = shift[ch];
  for (int i = blockIdx.x * blockDim.x + threadIdx.x; i < Np; i += gridDim.x * blockDim.x)
    X[base + i] = fmaxf(X[base + i] * s + sh, 0.f);
}

// 3x3 inverse of intrinsics (adjugate), one thread per batch element.
__global__ void prep_kinv_kernel(const float* __restrict__ K, float* __restrict__ kinv) {
  int b = threadIdx.x;
  if (b >= 12) return;
  const float* k = K + b * 9;
  float a = k[0], bb = k[1], c = k[2], d = k[3], e = k[4], f = k[5],
        g = k[6], h = k[7], i = k[8];
  float A = e * i - f * h, Bc = -(d * i - f * g), Cc = d * h - e * g;
  float inv = 1.0f / (a * A + bb * Bc + c * Cc);
  float* o = kinv + b * 9;
  o[0] = A * inv;               o[1] = -(bb * i - c * h) * inv; o[2] = (bb * f - c * e) * inv;
  o[3] = Bc * inv;              o[4] = (a * i - c * g) * inv;   o[5] = -(a * f - c * d) * inv;
  o[6] = Cc * inv;              o[7] = -(a * h - bb * g) * inv; o[8] = (a * e - bb * d) * inv;
}

// ---------------------------------------------------------------------------

extern "C" void kernel_launch(void* const* d_in, const int* in_sizes, int n_in,
                              void* d_out, int out_size, void* d_ws, size_t ws_size,
                              hipStream_t stream) {
  (void)out_size; (void)ws_size; (void)in_sizes;
  static const int SC[5] = {32, 64, 128, 256, 512};
  static const int SH[5] = {112, 56, 28, 14, 7};
  static const int SW[5] = {200, 100, 50, 25, 13};
  const int Bn = 12, NMAX = 22400, NB = 16384;

  const float* feat[5];
  for (int s = 0; s < 5; ++s) feat[s] = (const float*)d_in[s];
  const float* Kin = (const float*)d_in[5];
  const float* Tin = (const float*)d_in[6];

  // ---- parameter leaves, setup_inputs() insertion order:
  // per stage: fr_w1, fr_bn1{g,b,m,v}, fr_w2, fr_bn2{g,b,m,v},
  //            dn_w1, dn_bn1{g,b,m,v}, dn_w2, dn_b2,
  //            cf_w1, cf_bn1{g,b,m,v}, cf_w2, cf_b2   (24 leaves)
  // then fusion: w1, bn1{g,b,m,v}, w2, b2             (7 leaves)
  size_t lsz[127];
  int li = 0;
  for (int s = 0; s < 5; ++s) {
    size_t c = (size_t)SC[s];
    size_t t[24] = {64 * c, 64, 64, 64, 64, 9216, 128, 128, 128, 128,
                    64 * c, 64, 64, 64, 64, 4096, 64,
                    110592, 64, 64, 64, 64, 64, 1};
    for (int j = 0; j < 24; ++j) lsz[li++] = t[j];
  }
  {
    size_t t[7] = {163840, 256, 256, 256, 256, 32768, 128};
    for (int j = 0; j < 7; ++j) lsz[li++] = t[j];
  }
  const float* leaf[127];
  if (n_in >= 7 + 127) {
    for (int i = 0; i < 127; ++i) leaf[i] = (const float*)d_in[7 + i];
  } else {
    const float* base = (const float*)d_in[7];  // single concatenated buffer
    size_t off = 0;
    for (int i = 0; i < 127; ++i) { leaf[i] = base + off; off += lsz[i]; }
  }

  // ---- workspace carve-up (~580 MB of fp32, 256B aligned chunks)
  float* wsf = (float*)d_ws;
  size_t o = 0;
  auto alloc = [&](size_t nfl) {
    float* p = wsf + o;
    o += (nfl + 63) & ~(size_t)63;
    return p;
  };
  float* bnbuf = alloc(3712);                         // folded BN scale/shift
  float* kinv  = alloc(128);                          // 12 x 3x3 inverses
  float* hid   = alloc((size_t)Bn * 64 * NMAX);       // shared 64-ch hidden
  float* cat   = alloc((size_t)Bn * 192 * NMAX);      // [logits | reduced]
  float* dmap  = alloc((size_t)Bn * NMAX);
  float* cnf   = alloc((size_t)Bn * NMAX);
  float* bev   = alloc((size_t)Bn * 128 * NB);        // per-stage BEV
  float* pre   = alloc((size_t)Bn * 256 * NB);        // fusion pre-activation

  // ---- prep: fold BN params, invert intrinsics, zero accumulators
  for (int s = 0; s < 5; ++s) {
    int base = s * 24;
    float* sb = bnbuf + s * 640;
    fold_bn_kernel<<<1, 128, 0, stream>>>(leaf[base + 1], leaf[base + 2],
                                          leaf[base + 3], leaf[base + 4],
                                          sb + 0, sb + 64, 64);        // fr_bn1
    fold_bn_kernel<<<1, 128, 0, stream>>>(leaf[base + 6], leaf[base + 7],
                                          leaf[base + 8], leaf[base + 9],
                                          sb + 128, sb + 256, 128);    // fr_bn2
    fold_bn_kernel<<<1, 128, 0, stream>>>(leaf[base + 11], leaf[base + 12],
                                          leaf[base + 13], leaf[base + 14],
                                          sb + 384, sb + 448, 64);     // dn_bn1
    fold_bn_kernel<<<1, 128, 0, stream>>>(leaf[base + 18], leaf[base + 19],
                                          leaf[base + 20], leaf[base + 21],
                                          sb + 512, sb + 576, 64);     // cf_bn1
  }
  fold_bn_kernel<<<1, 256, 0, stream>>>(leaf[121], leaf[122], leaf[123], leaf[124],
                                        bnbuf + 3200, bnbuf + 3456, 256);  // fusion bn
  prep_kinv_kernel<<<1, 32, 0, stream>>>(Kin, kinv);
  hipMemsetAsync(pre, 0, (size_t)Bn * 256 * NB * sizeof(float), stream);

  const dim3 blk(32 * WPB);
  // MT=4 everywhere: grid.y covers M in chunks of 64 rows per wave.
  auto grd = [&](int M, int Nn) {
    return dim3((unsigned)((Nn + 16 * WPB - 1) / (16 * WPB)),
                (unsigned)((M + 63) / 64), (unsigned)Bn);
  };

  // ---- per-stage pipeline
  for (int s = 0; s < 5; ++s) {
    const int c = SC[s], H = SH[s], Wd = SW[s], N = H * Wd;
    const int base = s * 24;
    const float* sb = bnbuf + s * 640;

    // feature_reduction 1x1 (c->64) + BN + ReLU
    gemm16_kernel<4, 1, true, false><<<grd(64, N), blk, 0, stream>>>(
        leaf[base + 0], c, feat[s], (long long)c * N, N,
        hid, (long long)64 * N, N, sb + 0, sb + 64, 64, c, N);
    // feature_reduction 3x3 grouped(8) (64->128) + BN + ReLU -> cat[64:192]
    // Mpg=16 -> MT=1 (one 16-row tile per group)
    conv3x3_kernel<1, true><<<dim3((N + 63) / 64, 8, Bn), blk, 0, stream>>>(
        leaf[base + 5], hid, (long long)64 * N, cat, (long long)192 * N,
        sb + 128, sb + 256, H, Wd, 8, 16, 8, 64);
    // depth_net 1x1 (c->64) + BN + ReLU
    gemm16_kernel<4, 1, true, false><<<grd(64, N), blk, 0, stream>>>(
        leaf[base + 10], c, feat[s], (long long)c * N, N,
        hid, (long long)64 * N, N, sb + 384, sb + 448, 64, c, N);
    // depth_net 1x1 (64->64) + bias -> logits in cat[0:64]
    gemm16_kernel<4, 2, false, false><<<grd(64, N), blk, 0, stream>>>(
        leaf[base + 15], 64, hid, (long long)64 * N, N,
        cat, (long long)192 * N, N, nullptr, leaf[base + 16], 64, 64, N);
    // expected depth
    depth_softmax_kernel<<<dim3((N + 255) / 256, Bn), 256, 0, stream>>>(
        cat, (long long)192 * N, dmap, N);
    // confidence 3x3 (192->64) + BN + ReLU: Mpg=64, MT=4 -> gather shared by
    // all 64 output channels (grid.y = 1)
    conv3x3_kernel<4, true><<<dim3((N + 63) / 64, 1, Bn), blk, 0, stream>>>(
        leaf[base + 17], cat, (long long)192 * N, hid, (long long)64 * N,
        sb + 512, sb + 576, H, Wd, 192, 64, 1, 0);
    // confidence 1x1 (64->1) + sigmoid
    conf_kernel<<<dim3((N + 255) / 256, Bn), 256, 0, stream>>>(
        hid, (long long)64 * N, leaf[base + 22], leaf[base + 23], cnf, N);
    // BEV scatter-add
    hipMemsetAsync(bev, 0, (size_t)Bn * 128 * NB * sizeof(float), stream);
    scatter_kernel<<<dim3((N + 127) / 128, Bn), 128, 0, stream>>>(
        cat, (long long)192 * N, cnf, dmap, kinv, Tin, bev, H, Wd);
    // fusion partial:  pre256 += W1[:, s*128 : s*128+128] @ bev_s
    gemm16_kernel<4, 0, false, true><<<grd(256, NB), blk, 0, stream>>>(
        leaf[120] + s * 128, 640, bev, (long long)128 * NB, NB,
        pre, (long long)256 * NB, NB, nullptr, nullptr, 256, 128, NB);
  }

  // ---- fusion: ReLU(BN(pre256)), then 1x1 (256->128) + bias -> output
  bnrelu_kernel<<<dim3(64, 256, Bn), 256, 0, stream>>>(pre, bnbuf + 3200,
                                                       bnbuf + 3456, 256, NB);
  gemm16_kernel<4, 2, false, false><<<grd(128, NB), blk, 0, stream>>>(
      leaf[125], 256, pre, (long long)256 * NB, NB,
      (float*)d_out, (long long)128 * NB, NB, nullptr, leaf[126], 128, 256, NB);
}